// SubtitleColorConsistencyLoss_28157805593440
// MI455X (gfx1250) — compile-verified
//
#include <hip/hip_runtime.h>
#include <stdint.h>

#define NIMG 8
#define CCH  16
#define HH   512
#define WW   512
#define KSEG 65
#define ROWS 8
#define PIX  (ROWS*WW)            // 4096 pixels per band
#define TPB  256
#define RUN  (PIX/TPB)            // 16 consecutive pixels per lane
#define CPAD 4112                 // 4096 + swizzle padding (<=1026 chunks)
#define FS   17                   // padded fsum row stride (bank spread)
#define TABJ 18                   // per-segment stats: cnt, sqsum, fsum[16]
#define TABF (NIMG*KSEG*TABJ)     // 9360 floats global table

typedef __attribute__((ext_vector_type(2))) float v2f;
typedef __attribute__((ext_vector_type(8))) float v8f;

__device__ __forceinline__ uint32_t lds_off32(const void* p) {
  // generic pointer to LDS: low 32 bits are the group-segment byte offset
  return (uint32_t)(uintptr_t)p;
}

// bank-conflict-avoiding placement: 16B chunk g -> slot g + ((g>>4)&3)
__device__ __forceinline__ uint32_t slot_of(uint32_t g) {
  return g + ((g >> 4) & 3u);
}

// CDNA5 async copy: global -> LDS, 16B per lane, tracked by ASYNCcnt (GVS mode)
__device__ __forceinline__ void async_b128(uint32_t lds_byte, uint32_t voff, uint64_t sbase) {
  asm volatile("global_load_async_to_lds_b128 %0, %1, %2"
               :: "v"(lds_byte), "v"(voff), "s"(sbase) : "memory");
}

__global__ __launch_bounds__(TPB) void kzero(float* __restrict__ g) {
  int i = blockIdx.x * TPB + threadIdx.x;
  if (i < TABF) g[i] = 0.0f;
}

__global__ __launch_bounds__(TPB) void kmain(const float* __restrict__ color,
                                             const int*   __restrict__ labels,
                                             float*       __restrict__ gtab) {
  __shared__ int   s_lab[PIX];          // 16 KB (identity layout; read once)
  __shared__ float s_col[2][CPAD];      // ~32 KB double buffer (swizzled)
  __shared__ float s_cnt[KSEG];
  __shared__ float s_sq[KSEG];
  __shared__ float s_fs[KSEG*FS];

  const int t    = threadIdx.x;
  const int n    = blockIdx.x >> 6;     // 64 bands per image
  const int band = blockIdx.x & 63;
  const int row0 = band * ROWS;

  for (int i = t; i < KSEG; i += TPB) { s_cnt[i] = 0.f; s_sq[i] = 0.f; }
  for (int i = t; i < KSEG*FS; i += TPB) s_fs[i] = 0.f;

  // ---- issue async label tile (4 x b128 per lane, identity layout) ----
  const uint64_t lab_base = (uint64_t)(uintptr_t)(labels + ((size_t)n*HH + row0) * WW);
  const uint32_t lab_lds  = lds_off32(s_lab);
#pragma unroll
  for (int r = 0; r < 4; ++r) {
    uint32_t off = (uint32_t)(r*1024 + t*4) * 4u;
    async_b128(lab_lds + off, off, lab_base);
  }
  // ---- issue channel 0 (swizzled placement) ----
  {
    const uint64_t cb = (uint64_t)(uintptr_t)(color + (((size_t)n*CCH + 0)*HH + row0) * WW);
    const uint32_t cl = lds_off32(&s_col[0][0]);
#pragma unroll
    for (int r = 0; r < 4; ++r) {
      uint32_t g = (uint32_t)(r*256 + t);
      async_b128(cl + slot_of(g)*16u, g*16u, cb);
    }
  }
  asm volatile("s_wait_asynccnt 0x4" ::: "memory");  // labels done (loads in-order)
  __syncthreads();

  // per-lane run label (16 consecutive px; uniform for 64-aligned blocks)
  const int pbase  = t * RUN;
  const int runlab = s_lab[pbase];
  bool uniform = true;
#pragma unroll
  for (int i = 1; i < RUN; ++i) uniform &= (s_lab[pbase + i] == runlab);

  // lane's swizzled word base: 4 consecutive chunks starting at slot 4t + ((t>>2)&3)
  const int wbase = 16*t + 4*((t >> 2) & 3);

  for (int c = 0; c < CCH; ++c) {
    if (c < CCH - 1) {
      const uint64_t nb   = (uint64_t)(uintptr_t)(color + (((size_t)n*CCH + (c+1))*HH + row0) * WW);
      const uint32_t nlds = lds_off32(&s_col[(c+1)&1][0]);
#pragma unroll
      for (int r = 0; r < 4; ++r) {
        uint32_t g = (uint32_t)(r*256 + t);
        async_b128(nlds + slot_of(g)*16u, g*16u, nb);
      }
      asm volatile("s_wait_asynccnt 0x4" ::: "memory"); // current channel landed
    } else {
      asm volatile("s_wait_asynccnt 0x0" ::: "memory");
    }
    __syncthreads();

    const float* __restrict__ buf = s_col[c & 1];
    if (uniform) {
      float s = 0.f, q = 0.f;
#pragma unroll
      for (int i = 0; i < RUN; ++i) { float f = buf[wbase + i]; s += f; q += f * f; }
      atomicAdd(&s_fs[runlab*FS + c], s);
      atomicAdd(&s_sq[runlab], q);
      if (c == 0) atomicAdd(&s_cnt[runlab], (float)RUN);
    } else {
      for (int i = 0; i < RUN; ++i) {
        int   l = s_lab[pbase + i];
        float f = buf[wbase + i];
        atomicAdd(&s_fs[l*FS + c], f);
        atomicAdd(&s_sq[l], f * f);
        if (c == 0) atomicAdd(&s_cnt[l], 1.f);
      }
    }
    __syncthreads();
  }

  // flush band table -> global (skip zeros)
  for (int i = t; i < KSEG*TABJ; i += TPB) {
    int k = i / TABJ, j = i % TABJ;
    float v = (j == 0) ? s_cnt[k] : (j == 1) ? s_sq[k] : s_fs[k*FS + (j - 2)];
    if (v != 0.f) atomicAdd(&gtab[((size_t)n*KSEG + k)*TABJ + j], v);
  }
}

// Gram diagonal of 16x16 row-block X via V_WMMA_F32_16X16X4_F32 (A == B == X rows)
__device__ __forceinline__ float gram_diag(int w, int m, bool hi,
                                           float (*X)[16][FS], float (*D)[16][FS]) {
  v8f acc = {};
#pragma unroll
  for (int s = 0; s < 4; ++s) {
    v2f a;
    a.x = X[w][m][4*s + (hi ? 2 : 0)];
    a.y = X[w][m][4*s + (hi ? 3 : 1)];
    acc = __builtin_amdgcn_wmma_f32_16x16x4_f32(false, a, false, a, (short)0, acc, false, false);
  }
#pragma unroll
  for (int r = 0; r < 8; ++r) {
    int rowm = r + (hi ? 8 : 0);
    D[w][rowm][m] = acc[r];
  }
  asm volatile("s_wait_dscnt 0x0" ::: "memory");
  return hi ? 0.f : D[w][m][m];
}

__global__ __launch_bounds__(TPB) void kfinal(const float* __restrict__ gtab,
                                              float* __restrict__ out) {
  __shared__ float s_mbg[NIMG][CCH];
  __shared__ float s_bgc[NIMG];
  __shared__ float s_X[8][16][FS];
  __shared__ float s_D[8][16][FS];
  __shared__ float s_red[3][TPB];

  const int  t    = threadIdx.x;
  const int  w    = t >> 5;
  const int  lane = t & 31;
  const int  m    = lane & 15;
  const bool hi   = lane >= 16;

  if (t < NIMG*CCH) {
    int n = t >> 4, c = t & 15;
    const float* r0 = gtab + (size_t)n*KSEG*TABJ;   // segment (n, k=0) = background
    s_mbg[n][c] = r0[2 + c] / (r0[0] + 1e-6f);
    if (c == 0) s_bgc[n] = r0[0];
  }
  __syncthreads();

  float vsum = 0.f, isum = 0.f, esum = 0.f;
  const int TILES  = (NIMG*KSEG + 15) / 16;   // 33
  const int ROUNDS = (TILES + 7) / 8;         // 5

  for (int r = 0; r < ROUNDS; ++r) {
    int  tile  = r*8 + w;
    int  seg   = tile*16 + m;
    bool segok = (tile < TILES) && (seg < NIMG*KSEG);
    int  segc  = segok ? seg : 0;
    const float* row = gtab + (size_t)segc * TABJ;
    int n = segc / KSEG, k = segc % KSEG;

    float cnt = segok ? row[0] : 0.f;
    float sq  = segok ? row[1] : 0.f;
    float denom = cnt + 1e-6f;

    if (!hi) {
#pragma unroll
      for (int c = 0; c < CCH; ++c) s_X[w][m][c] = segok ? row[2 + c] : 0.f;
    }
    asm volatile("s_wait_dscnt 0x0" ::: "memory");
    float qf = gram_diag(w, m, hi, s_X, s_D);          // ||fsum||^2

    if (!hi) {
#pragma unroll
      for (int c = 0; c < CCH; ++c) {
        float fs = segok ? row[2 + c] : 0.f;
        s_X[w][m][c] = segok ? (fs / denom - s_mbg[n][c]) : 0.f;
      }
    }
    asm volatile("s_wait_dscnt 0x0" ::: "memory");
    float qd = gram_diag(w, m, hi, s_X, s_D);          // ||m_sub - m_bg||^2

    if (!hi && segok) {
      float intra = (sq - 2.f*qf/denom + cnt*qf/(denom*denom)) / denom;
      float dist  = sqrtf(qd + 1e-12f);
      float md    = 0.5f - dist;
      float inter = md > 0.f ? md*md : 0.f;
      bool  valid = (cnt >= 20.f) && (k > 0) && (s_bgc[n] >= 20.f);
      if (valid) { vsum += 1.f; isum += intra; esum += inter; }
    }
  }

  s_red[0][t] = vsum; s_red[1][t] = isum; s_red[2][t] = esum;
  __syncthreads();
  for (int st = TPB/2; st > 0; st >>= 1) {
    if (t < st) {
      s_red[0][t] += s_red[0][t+st];
      s_red[1][t] += s_red[1][t+st];
      s_red[2][t] += s_red[2][t+st];
    }
    __syncthreads();
  }
  if (t == 0) {
    float total = s_red[0][0];
    float safe  = total > 1.f ? total : 1.f;
    out[0] = (total > 0.f) ? (s_red[1][0]/safe + 0.4f*s_red[2][0]/safe) : 0.f;
  }
}

extern "C" void kernel_launch(void* const* d_in, const int* in_sizes, int n_in,
                              void* d_out, int out_size, void* d_ws, size_t ws_size,
                              hipStream_t stream) {
  const float* color  = (const float*)d_in[0];
  // d_in[1] = gt (float): redundant — labels==0 exactly where gt<=0.5
  const int*   labels = (const int*)d_in[2];
  float* gtab = (float*)d_ws;

  kzero <<<(TABF + TPB - 1)/TPB, TPB, 0, stream>>>(gtab);
  kmain <<<NIMG * (HH / ROWS), TPB, 0, stream>>>(color, labels, gtab);
  kfinal<<<1, TPB, 0, stream>>>(gtab, (float*)d_out);
}